// MOE_86328842649680
// MI455X (gfx1250) — compile-verified
//
#include <hip/hip_runtime.h>
#include <math.h>

typedef float v2f __attribute__((ext_vector_type(2)));
typedef float v8f __attribute__((ext_vector_type(8)));
typedef unsigned int v4u __attribute__((ext_vector_type(4)));
typedef int v4i __attribute__((ext_vector_type(4)));
typedef int v8i __attribute__((ext_vector_type(8)));

#define BB 4
#define SS 2048
#define DD 1024
#define EE 8
#define NN 16
#define RR 4

#define KC 64              // K-chunk staged per TDM op
#define NCH (DD / KC)      // 16 chunks
#define LROW (KC + 4)      // 68 floats: TDM pad 4 DWORDs per 64 -> conflict-free

// ---------------------------------------------------------------------------
// TDM: issue a 2-D tile load (rows x KC f32, row stride = strideElems) into
// LDS at byte offset lds_off, with 4-DWORD padding after every 64 DWORDs.
// Descriptor per CDNA5 ISA ch.8 (D# group0 + group1; groups 2/3 unused).
// ---------------------------------------------------------------------------
__device__ __forceinline__ void tdm_load_2d(unsigned lds_byte_off,
                                            const float* gptr,
                                            unsigned tile_d0,   // elems per row
                                            unsigned tile_d1,   // rows
                                            unsigned strideElems) {
  unsigned long long ga = (unsigned long long)(uintptr_t)gptr;
  v4u g0;
  g0.x = 1u;                                   // count=1 (valid), user mode
  g0.y = lds_byte_off;                         // lds_addr
  g0.z = (unsigned)(ga & 0xffffffffull);       // global_addr[31:0]
  g0.w = (unsigned)((ga >> 32) & 0x1ffffffull) // global_addr[56:32]
         | (2u << 30);                         // type=2 ("image")
  v8i g1;
  g1[0] = (int)((2u << 16)      // data_size = 4 bytes
                | (1u << 20)    // pad_enable
                | (5u << 22)    // pad_interval: 64 DWORDs
                | (3u << 25));  // pad_amount:   4 DWORDs
  g1[1] = (int)(((unsigned)DD & 0xffffu) << 16);            // tensor_dim0 lo16
  g1[2] = (int)((((unsigned)DD >> 16) & 0xffffu)            // tensor_dim0 hi16
                | ((tile_d1 & 0xffffu) << 16));             // tensor_dim1 lo16
  g1[3] = (int)((tile_d0 & 0xffffu) << 16);                 // dim1 hi=0 | tile_dim0
  g1[4] = (int)(tile_d1 & 0xffffu);                         // tile_dim1 | tile_dim2=0
  g1[5] = (int)strideElems;                                 // tensor_dim0_stride lo32
  g1[6] = 0;                                                // stride hi16 | dim1_stride lo
  g1[7] = 0;
  v4i z4 = {0, 0, 0, 0};
#if __clang_major__ >= 23
  v8i z8 = {0, 0, 0, 0, 0, 0, 0, 0};
  __builtin_amdgcn_tensor_load_to_lds(g0, g1, z4, z4, z8, 0);
#else
  __builtin_amdgcn_tensor_load_to_lds(g0, g1, z4, z4, 0);
#endif
}

__device__ __forceinline__ unsigned lds_off_of(const void* p) {
  // generic pointer to LDS: low 32 bits are the LDS byte offset
  return (unsigned)(unsigned long long)(uintptr_t)p;
}

// ---------------------------------------------------------------------------
// K1: raw gate logits  raw[b,s,e] = x[b,s,:] . Wg[e,:]   (one wave per token)
// ---------------------------------------------------------------------------
__global__ __launch_bounds__(256) void k_logits(const float* __restrict__ x,
                                                const float* __restrict__ Wg,
                                                float* __restrict__ raw) {
  const int wave = blockIdx.x * 8 + (threadIdx.x >> 5);
  const int lane = threadIdx.x & 31;
  const float* xr = x + (size_t)wave * DD;
  float acc[EE];
#pragma unroll
  for (int e = 0; e < EE; ++e) acc[e] = 0.f;
  for (int d = lane; d < DD; d += 32) {
    float xv = xr[d];
#pragma unroll
    for (int e = 0; e < EE; ++e) acc[e] += xv * Wg[e * DD + d];
  }
#pragma unroll
  for (int e = 0; e < EE; ++e) {
    float v = acc[e];
    for (int m = 16; m >= 1; m >>= 1) v += __shfl_xor(v, m, 32);
    if (lane == 0) raw[(size_t)wave * EE + e] = v;
  }
}

// ---------------------------------------------------------------------------
// K2: nk[r,b,n,d] = LN(ref[b,n,:]) * lnk_s[r,:] + lnk_b[r,:]   (1 MB total)
// ---------------------------------------------------------------------------
__global__ __launch_bounds__(256) void k_nk(const float* __restrict__ ref,
                                            const float* __restrict__ lnk_s,
                                            const float* __restrict__ lnk_b,
                                            float* __restrict__ nk) {
  __shared__ float red[16];
  const int row = blockIdx.x;  // b*NN + n
  const int tid = threadIdx.x;
  const float* rr = ref + (size_t)row * DD;
  float s = 0.f, ss = 0.f;
  for (int d = tid; d < DD; d += 256) {
    float v = rr[d];
    s += v;
    ss += v * v;
  }
  for (int m = 16; m >= 1; m >>= 1) {
    s += __shfl_xor(s, m, 32);
    ss += __shfl_xor(ss, m, 32);
  }
  const int wv = tid >> 5, ln = tid & 31;
  if (ln == 0) {
    red[wv] = s;
    red[8 + wv] = ss;
  }
  __syncthreads();
  if (tid == 0) {
    float ts = 0.f, tss = 0.f;
    for (int i = 0; i < 8; ++i) {
      ts += red[i];
      tss += red[8 + i];
    }
    red[0] = ts;
    red[8] = tss;
  }
  __syncthreads();
  const float mu = red[0] / (float)DD;
  const float var = red[8] / (float)DD - mu * mu;
  const float rstd = rsqrtf(var + 1e-5f);
  const int b = row / NN, n = row % NN;
  for (int d = tid; d < DD; d += 256) {
    float nr = (rr[d] - mu) * rstd;
#pragma unroll
    for (int r = 0; r < RR; ++r)
      nk[((size_t)(r * BB + b) * NN + n) * DD + d] =
          nr * lnk_s[r * DD + d] + lnk_b[r * DD + d];
  }
}

// ---------------------------------------------------------------------------
// K3: fused main kernel. 128 threads (4 waves) per 16-token tile.
// TDM double-buffers x and nk K-chunks into LDS; WMMA consumes from LDS.
// ---------------------------------------------------------------------------
__global__ __launch_bounds__(128) void k_main(
    const float* __restrict__ x, const float* __restrict__ Wmap,
    const float* __restrict__ lnq_s, const float* __restrict__ lnq_b,
    const float* __restrict__ const_c, const float* __restrict__ const_w,
    const float* __restrict__ raw, const float* __restrict__ nk,
    float* __restrict__ out, float* __restrict__ logits_out) {
  __shared__ __align__(16) float s_x[2][16][LROW];       //  8.5 KB
  __shared__ __align__(16) float s_nk[2][RR][16][LROW];  // 34   KB
  __shared__ float s_mu[16], s_rstd[16], s_cl[16][4];
  __shared__ float s_alpha[16], s_beta[16][2], s_wv[16][4];
  __shared__ float s_attn[RR][16 * 17];

  const int tid = threadIdx.x;
  const int wave = tid >> 5;
  const int lane = tid & 31;
  const int tilesPerB = SS / 16;
  const int b = blockIdx.x / tilesPerB;
  const int s0 = (blockIdx.x % tilesPerB) * 16;
  const size_t xbase = ((size_t)b * SS + s0) * DD;
  const float* nkr = nk + ((size_t)(wave * BB + b) * NN) * DD;

  // ---- TDM prologue: prefetch chunks 0 and 1 (overlaps phases 1-2) ----
#pragma unroll
  for (int c = 0; c < 2; ++c) {
    tdm_load_2d(lds_off_of(&s_nk[c][wave][0][0]), nkr + c * KC, KC, NN, DD);
    if (wave == 0)
      tdm_load_2d(lds_off_of(&s_x[c][0][0]), x + xbase + c * KC, KC, 16, DD);
  }

  // ---- Phase 1: per-token mean/var of xs=2x and const-cluster dots ----
  {
    const int m = tid >> 3;  // 0..15
    const int sub = tid & 7;
    const float* xr = x + xbase + (size_t)m * DD;
    float s = 0.f, ssum = 0.f, c00 = 0.f, c01 = 0.f, c10 = 0.f, c11 = 0.f;
    for (int d = sub; d < DD; d += 8) {
      float xs = 2.f * xr[d];
      s += xs;
      ssum += xs * xs;
      c00 += xs * const_w[0 * DD + d];
      c01 += xs * const_w[1 * DD + d];
      c10 += xs * const_w[2 * DD + d];
      c11 += xs * const_w[3 * DD + d];
    }
#pragma unroll
    for (int msk = 4; msk >= 1; msk >>= 1) {
      s += __shfl_xor(s, msk, 8);
      ssum += __shfl_xor(ssum, msk, 8);
      c00 += __shfl_xor(c00, msk, 8);
      c01 += __shfl_xor(c01, msk, 8);
      c10 += __shfl_xor(c10, msk, 8);
      c11 += __shfl_xor(c11, msk, 8);
    }
    if (sub == 0) {
      float mu = s / (float)DD;
      float var = ssum / (float)DD - mu * mu;
      s_mu[m] = mu;
      s_rstd[m] = rsqrtf(var + 1e-5f);
      s_cl[m][0] = c00;
      s_cl[m][1] = c01;
      s_cl[m][2] = c10;
      s_cl[m][3] = c11;
    }
  }
  __syncthreads();

  // ---- Phase 2: gating + final logits (wave 0, lanes 0..15) ----
  if (wave == 0 && lane < 16) {
    const int m = lane;
    const int sg = s0 + m;
    float f[EE];
    const float* cur = raw + ((size_t)b * SS + sg) * EE;
#pragma unroll
    for (int e = 0; e < EE; ++e) f[e] = cur[e];
    if (sg > 0) {
      const float* prv = cur - EE;
#pragma unroll
      for (int fo = 0; fo < EE; ++fo) {
        float a = 0.f;
#pragma unroll
        for (int e = 0; e < EE; ++e) a += prv[e] * Wmap[fo * EE + e];
        f[fo] += a;
      }
    }
    float* lo = logits_out + ((size_t)b * SS + sg) * EE;
#pragma unroll
    for (int e = 0; e < EE; ++e) lo[e] = f[e];
    float mx = f[0];
#pragma unroll
    for (int e = 1; e < EE; ++e) mx = fmaxf(mx, f[e]);
    float g[EE], gs = 0.f;
#pragma unroll
    for (int e = 0; e < EE; ++e) {
      g[e] = expf(f[e] - mx);
      gs += g[e];
    }
#pragma unroll
    for (int e = 0; e < EE; ++e) g[e] /= gs;
    int i1 = 0;
    float g1 = g[0];
#pragma unroll
    for (int e = 1; e < EE; ++e)
      if (g[e] > g1) { g1 = g[e]; i1 = e; }
    int i2 = (i1 == 0) ? 1 : 0;
    float g2 = g[i2];
#pragma unroll
    for (int e = 0; e < EE; ++e)
      if (e != i1 && g[e] > g2) { g2 = g[e]; i2 = e; }
    float t1 = (i1 == EE - 1) ? 0.f : g1;
    float t2 = (i2 == EE - 1) ? 0.f : g2;
    float inv = 1.f / (t1 + t2);
    float wv[EE];
#pragma unroll
    for (int e = 0; e < EE; ++e) wv[e] = 0.f;
    wv[i1] = t1 * inv;
    wv[i2] += t2 * inv;
    float a0 = s_cl[m][0], a1 = s_cl[m][1];
    float mxa = fmaxf(a0, a1);
    float e0 = expf(a0 - mxa), e1 = expf(a1 - mxa), tt = e0 + e1;
    float p00 = e0 / tt, p01 = e1 / tt;
    a0 = s_cl[m][2];
    a1 = s_cl[m][3];
    mxa = fmaxf(a0, a1);
    e0 = expf(a0 - mxa);
    e1 = expf(a1 - mxa);
    tt = e0 + e1;
    float p10 = e0 / tt, p11 = e1 / tt;

    s_alpha[m] = wv[0] + wv[1] + wv[2] + wv[3] + wv[4] * p00 + wv[5] * p10 + wv[6];
    s_beta[m][0] = wv[4] * p01;
    s_beta[m][1] = wv[5] * p11;
#pragma unroll
    for (int r = 0; r < RR; ++r) s_wv[m][r] = wv[r];
  }
  __syncthreads();

  // ---- Phase 3: scores via WMMA, A/B fragments from TDM-staged LDS ----
  {
    const int m = lane & 15;
    const int half = lane >> 4;
    const float mu = s_mu[m];
    const float rstd = s_rstd[m];
    const float* ls = lnq_s + wave * DD;
    const float* lb = lnq_b + wave * DD;
    v8f c = {};
    for (int ch = 0; ch < NCH; ++ch) {
      const int buf = ch & 1;
      // wait for this chunk's own-wave TDM (chunk ch+1 may stay in flight)
      if (ch + 1 < NCH) {
        if (wave == 0)
          __builtin_amdgcn_s_wait_tensorcnt(2);
        else
          __builtin_amdgcn_s_wait_tensorcnt(1);
      } else {
        __builtin_amdgcn_s_wait_tensorcnt(0);
      }
      __syncthreads();  // wave0's x-chunk visible to all waves
      const int k0 = ch * KC;
#pragma unroll
      for (int kk = 0; kk < KC; kk += 4) {
        const int ka = kk + 2 * half;
        v2f a, bfrag;
        float x0 = s_x[buf][m][ka];
        float x1 = s_x[buf][m][ka + 1];
        a.x = ((2.f * x0 - mu) * rstd) * ls[k0 + ka] + lb[k0 + ka];
        a.y = ((2.f * x1 - mu) * rstd) * ls[k0 + ka + 1] + lb[k0 + ka + 1];
        bfrag.x = s_nk[buf][wave][m][ka];
        bfrag.y = s_nk[buf][wave][m][ka + 1];
        c = __builtin_amdgcn_wmma_f32_16x16x4_f32(false, a, false, bfrag,
                                                  (short)0, c, false, false);
      }
      __syncthreads();  // everyone done with buf before it is overwritten
      if (ch + 2 < NCH) {
        tdm_load_2d(lds_off_of(&s_nk[buf][wave][0][0]), nkr + (ch + 2) * KC,
                    KC, NN, DD);
        if (wave == 0)
          tdm_load_2d(lds_off_of(&s_x[buf][0][0]), x + xbase + (ch + 2) * KC,
                      KC, 16, DD);
      }
    }
    // row softmax over n (16-lane groups), weight by w_r[m], store to LDS
#pragma unroll
    for (int j = 0; j < 8; ++j) {
      float v = c[j];
      float mx = v;
      mx = fmaxf(mx, __shfl_xor(mx, 1, 16));
      mx = fmaxf(mx, __shfl_xor(mx, 2, 16));
      mx = fmaxf(mx, __shfl_xor(mx, 4, 16));
      mx = fmaxf(mx, __shfl_xor(mx, 8, 16));
      float e = expf(v - mx);
      float sm = e;
      sm += __shfl_xor(sm, 1, 16);
      sm += __shfl_xor(sm, 2, 16);
      sm += __shfl_xor(sm, 4, 16);
      sm += __shfl_xor(sm, 8, 16);
      const int row = j + 8 * half;
      s_attn[wave][row * 17 + (lane & 15)] = (e / sm) * s_wv[row][wave];
    }
  }
  __syncthreads();

  // ---- Phase 5: out tile. Each wave owns a 256-column d-quarter.
  // G = sum_r (w_r * attn_r) @ nk_r accumulates in one WMMA C chain.
  {
    const int half = lane >> 4;
    const int nidx = lane & 15;
    const int dbase = wave * 256;
    const float* cc0 = const_c;
    const float* cc1 = const_c + DD;
    for (int t = 0; t < 16; ++t) {
      const int d0 = dbase + t * 16;
      const int d = d0 + nidx;
      v8f acc = {};
#pragma unroll
      for (int r = 0; r < RR; ++r) {
        const float* nkb = nk + ((size_t)(r * BB + b) * NN) * DD + d;
#pragma unroll
        for (int kk = 0; kk < 4; ++kk) {
          const int krow = 4 * kk + 2 * half;
          v2f a, bfrag;
          a.x = s_attn[r][nidx * 17 + krow];
          a.y = s_attn[r][nidx * 17 + krow + 1];
          bfrag.x = nkb[(size_t)krow * DD];
          bfrag.y = nkb[(size_t)(krow + 1) * DD];
          acc = __builtin_amdgcn_wmma_f32_16x16x4_f32(false, a, false, bfrag,
                                                      (short)0, acc, false,
                                                      false);
        }
      }
#pragma unroll
      for (int j = 0; j < 8; ++j) {
        const int m = j + 8 * half;
        const float xs = 2.f * x[xbase + (size_t)m * DD + d];
        float val = acc[j] + s_alpha[m] * xs + s_beta[m][0] * cc0[d] +
                    s_beta[m][1] * cc1[d];
        out[xbase + (size_t)m * DD + d] = val;
      }
    }
  }
}

// ---------------------------------------------------------------------------
extern "C" void kernel_launch(void* const* d_in, const int* in_sizes, int n_in,
                              void* d_out, int out_size, void* d_ws,
                              size_t ws_size, hipStream_t stream) {
  (void)in_sizes;
  (void)n_in;
  (void)out_size;
  (void)ws_size;
  const float* x = (const float*)d_in[0];
  const float* Wg = (const float*)d_in[1];
  const float* Wmap = (const float*)d_in[2];
  const float* lnq_s = (const float*)d_in[3];
  const float* lnq_b = (const float*)d_in[4];
  const float* lnk_s = (const float*)d_in[5];
  const float* lnk_b = (const float*)d_in[6];
  const float* ref = (const float*)d_in[7];
  const float* const_c = (const float*)d_in[8];
  const float* const_w = (const float*)d_in[9];
  float* out = (float*)d_out;
  float* logits_out = out + (size_t)BB * SS * DD;
  float* raw = (float*)d_ws;                  // B*S*E floats   (256 KB)
  float* nkbuf = raw + (size_t)BB * SS * EE;  // R*B*N*D floats (1 MB)

  k_logits<<<BB * SS / 8, 256, 0, stream>>>(x, Wg, raw);
  k_nk<<<BB * NN, 256, 0, stream>>>(ref, lnk_s, lnk_b, nkbuf);
  k_main<<<BB * (SS / 16), 128, 0, stream>>>(x, Wmap, lnq_s, lnq_b, const_c,
                                             const_w, raw, nkbuf, out,
                                             logits_out);
}